// Gcl_72215580115692
// MI455X (gfx1250) — compile-verified
//
#include <hip/hip_runtime.h>

typedef __attribute__((ext_vector_type(2))) float v2f;
typedef __attribute__((ext_vector_type(8))) float v8f;

#define N_NODES 100000
#define N_EDGES 3200000
#define N_REL   4
#define D_IN    256
#define D_OUT   128

// ---------- Phase 0: out[n][d] = bias[d] ----------
__global__ void init_bias_kernel(float* __restrict__ out,
                                 const float* __restrict__ bias) {
  size_t i = (size_t)blockIdx.x * blockDim.x + threadIdx.x;
  if (i < (size_t)N_NODES * D_OUT) out[i] = bias[i & (D_OUT - 1)];
}

// ---------- Phase 1: xw = x @ w[r] via V_WMMA_F32_16X16X4_F32 ----------
// One wave computes a 16(M) x 128(N) strip: 8 C-tiles of 16x16, K-loop over 256.
// A layout (16x4 f32): lane 0-15 -> M, vgpr v holds K = k + v + 2*(lane>>4)
// B layout (4x16 f32): lane%16 -> N, vgpr v holds K-row = k + v + 2*(lane>>4)
// C layout (16x16 f32): vgpr v -> row v + 8*(lane>>4), col = lane%16
__global__ void gemm_kernel(const float* __restrict__ x,
                            const float* __restrict__ w,   // pre-offset to relation r
                            float* __restrict__ xw) {
  const int lane = threadIdx.x & 31;
  const int wave = threadIdx.x >> 5;
  const int mtile = blockIdx.x * 4 + wave;
  if (mtile >= N_NODES / 16) return;           // wave-uniform guard (EXEC stays all-1s)

  const int half = lane >> 4;                   // 0 or 1
  const int col  = lane & 15;
  const int mrow = mtile * 16 + col;            // lane%16 = M row within tile
  const float* xp = x + (size_t)mrow * D_IN + 2 * half;

  v8f acc[8] = {};
  for (int k = 0; k < D_IN; k += 4) {
    const float2 av = *(const float2*)(xp + k); // K = k+2*half, k+2*half+1
    v2f a; a[0] = av.x; a[1] = av.y;
    const float* wrow = w + (size_t)(k + 2 * half) * D_OUT + col;
#pragma unroll
    for (int nt = 0; nt < 8; ++nt) {
      v2f b;
      b[0] = wrow[nt * 16];                     // K-row k+2*half
      b[1] = wrow[nt * 16 + D_OUT];             // K-row k+2*half+1
      acc[nt] = __builtin_amdgcn_wmma_f32_16x16x4_f32(
          /*neg_a=*/false, a, /*neg_b=*/false, b,
          /*c_mod=*/(short)0, acc[nt], /*reuse_a=*/false, /*reuse_b=*/false);
    }
  }

  float* op = xw + (size_t)mtile * 16 * D_OUT;
#pragma unroll
  for (int v = 0; v < 8; ++v) {
    const int mm = v + 8 * half;
#pragma unroll
    for (int nt = 0; nt < 8; ++nt)
      op[(size_t)mm * D_OUT + nt * 16 + col] = acc[nt][v];
  }
}

// ---------- Phase 2: out[row] += val * xw[col]  (one wave per edge) ----------
__global__ void spmm_kernel(const float* __restrict__ xw,
                            const int* __restrict__ rows,   // pre-offset to relation r
                            const int* __restrict__ cols,
                            const float* __restrict__ vals,
                            float* __restrict__ out) {
  const int lane = threadIdx.x & 31;
  const int wave = threadIdx.x >> 5;
  const long long e = (long long)blockIdx.x * 8 + wave;
  if (e >= N_EDGES) return;                    // wave-uniform

  // wave-uniform offsets -> scalar loads
  const int   row = rows[e];
  const int   cn  = cols[e];
  const float val = vals[e];

  // 32 lanes x float4 = full 512B row, fully coalesced; L2-resident per relation
  const float4 v = *(const float4*)(xw + (size_t)cn * D_OUT + lane * 4);
  float* op = out + (size_t)row * D_OUT + lane * 4;
  unsafeAtomicAdd(op + 0, val * v.x);
  unsafeAtomicAdd(op + 1, val * v.y);
  unsafeAtomicAdd(op + 2, val * v.z);
  unsafeAtomicAdd(op + 3, val * v.w);
}

extern "C" void kernel_launch(void* const* d_in, const int* in_sizes, int n_in,
                              void* d_out, int out_size, void* d_ws, size_t ws_size,
                              hipStream_t stream) {
  const float* x    = (const float*)d_in[0];
  const float* w    = (const float*)d_in[1];
  const float* bias = (const float*)d_in[2];
  const int*   rows = (const int*)d_in[3];
  const int*   cols = (const int*)d_in[4];
  const float* vals = (const float*)d_in[5];
  float* out = (float*)d_out;
  float* xw  = (float*)d_ws;                   // 100000*128 floats = 51.2 MB scratch

  // out = broadcast(bias)
  init_bias_kernel<<<(N_NODES * D_OUT + 255) / 256, 256, 0, stream>>>(out, bias);

  // Relation-by-relation: keeps xw[r] (51.2 MB) hot in the 192 MB L2 for the SpMM gather.
  const int gemm_blocks = (N_NODES / 16 + 3) / 4;      // 4 waves/block, 16 rows/wave
  const int spmm_blocks = (N_EDGES + 7) / 8;           // 8 waves/block, 1 edge/wave
  for (int r = 0; r < N_REL; ++r) {
    gemm_kernel<<<gemm_blocks, 128, 0, stream>>>(
        x, w + (size_t)r * D_IN * D_OUT, xw);
    spmm_kernel<<<spmm_blocks, 256, 0, stream>>>(
        xw, rows + (size_t)r * N_EDGES, cols + (size_t)r * N_EDGES,
        vals + (size_t)r * N_EDGES, out);
  }
}